// VarDecoder_14559939134102
// MI455X (gfx1250) — compile-verified
//
#include <hip/hip_runtime.h>
#include <hip/hip_bf16.h>

typedef __attribute__((ext_vector_type(2))) float v2f;
typedef __attribute__((ext_vector_type(8))) float v8f;
typedef __attribute__((ext_vector_type(4))) unsigned int v4u;
typedef __attribute__((ext_vector_type(4))) int v4i;
typedef __attribute__((ext_vector_type(8))) int v8i;

// ---------------------------------------------------------------------------
// TDM stage: DMA the block's 16-row A stripe (16 x K f32, row-major) into LDS.
// D# built per CDNA5 ISA 8.3-8.5:
//   group0: count=1 | lds_addr | global_addr | type=2
//   group1: data_size=4B, tensor_dim0=K, tensor_dim1=16, tile 16xK, stride K
// Issued by wave 0 only (TDM ignores EXEC; one DMA per block), waited with
// s_wait_tensorcnt 0, then released to all waves via the workgroup barrier.
// ---------------------------------------------------------------------------
__device__ __forceinline__
void tdm_load_a_stripe(const float* gA, float* ldsA, int K) {
#if __has_builtin(__builtin_amdgcn_tensor_load_to_lds)
    if ((threadIdx.x >> 5) == 0) {
        unsigned long long ga = (unsigned long long)(const void*)gA;
        unsigned lds_off = (unsigned)(unsigned long long)ldsA;

        v4u g0;
        g0.x = 1u;                                          // count=1, user mode
        g0.y = lds_off;                                     // lds_addr
        g0.z = (unsigned)(ga & 0xFFFFFFFFull);              // global_addr[31:0]
        g0.w = (unsigned)((ga >> 32) & 0x01FFFFFFull)       // global_addr[56:32]
             | (2u << 30);                                  // type=2 (image)

        unsigned w0 = 2u << 16;                             // data_size=2 -> 4B
        unsigned w1 = ((unsigned)K & 0xFFFFu) << 16;        // tensor_dim0[15:0]
        unsigned w2 = (((unsigned)K >> 16) & 0xFFFFu)       // tensor_dim0[31:16]
                    | (16u << 16);                          // tensor_dim1[15:0]=16
        unsigned w3 = ((unsigned)K & 0xFFFFu) << 16;        // tile_dim0 = K
        unsigned w4 = 16u;                                  // tile_dim1 = 16
        unsigned w5 = (unsigned)K;                          // dim0_stride[31:0] = K
        v8i g1 = { (int)w0, (int)w1, (int)w2, (int)w3,
                   (int)w4, (int)w5, 0, 0 };
        v4i z4 = { 0, 0, 0, 0 };
#if __clang_major__ >= 23
        v8i z8 = { 0, 0, 0, 0, 0, 0, 0, 0 };
        __builtin_amdgcn_tensor_load_to_lds(g0, g1, z4, z4, z8, 0);
#else
        __builtin_amdgcn_tensor_load_to_lds(g0, g1, z4, z4, 0);
#endif
        __builtin_amdgcn_s_wait_tensorcnt(0);
    }
    __syncthreads();
#else
    // Fallback: cooperative fill, same [16][K] row-major LDS layout.
    for (int idx = threadIdx.x; idx < 16 * K; idx += blockDim.x)
        ldsA[idx] = gA[(idx / K) * (long)K + (idx % K)];
    __syncthreads();
#endif
}

// ---------------------------------------------------------------------------
// C[M,N] = A[M,K] @ B[K,N] + bias[N]   (all f32, row-major)
// Each wave computes a 16x64 tile with V_WMMA_F32_16X16X4_F32:
//   4 accumulators (v8f each); A fragment comes from the TDM-staged LDS
//   stripe (shared by all 8 waves -> 8x less global A traffic).
// Block = 256 threads = 8 waves -> block covers 16 rows x 512 cols.
// grid = (ceil(N/512), M/16); dynamic LDS = 16*K*4 bytes.
// permute_rows: hs row t = s*256 + v maps to output row v*8 + s, and the
// epilogue uses non-temporal stores (streaming 262 MB output must not rinse
// the L2-resident Wl).
// ---------------------------------------------------------------------------
__global__ __launch_bounds__(256)
void wmma_gemm_bias_f32(const float* __restrict__ A,
                        const float* __restrict__ B,
                        const float* __restrict__ bias,
                        float* __restrict__ C,
                        int M, int N, int K, int permute_rows) {
    extern __shared__ float Atile[];     // [16][K] row-major
    const int lane = threadIdx.x & 31;
    const int wave = threadIdx.x >> 5;
    const int row0 = blockIdx.y * 16;
    const int col0 = blockIdx.x * 512 + wave * 64;

    // Stage A stripe for the whole block (before any wave can drop out:
    // all waves must reach the barrier inside).
    tdm_load_a_stripe(A + (long)row0 * K, Atile, K);

    if (col0 >= N) return;               // wave-uniform tail drop

    // A fragment: lane L holds A[(L&15)][k + 2*(L>>4) + {0,1}] from LDS
    const int m  = lane & 15;
    const int kp = (lane >> 4) << 1;     // 0 or 2
    const float* Arow = Atile + (long)m * K;

    v8f acc0 = {}, acc1 = {}, acc2 = {}, acc3 = {};

    for (int k = 0; k < K; k += 4) {
        v2f a = *(const v2f*)(Arow + k + kp);   // ds_load_b64

        // B fragment (N striped across lanes):
        // lane L holds B[k + 2*(L>>4) + {0,1}][col0 + (L&15) + {0,16,32,48}]
        const float* B0 = B + (long)(k + kp)     * N + col0 + m;
        const float* B1 = B + (long)(k + kp + 1) * N + col0 + m;
        v2f b0, b1, b2, b3;
        b0.x = B0[0];   b0.y = B1[0];
        b1.x = B0[16];  b1.y = B1[16];
        b2.x = B0[32];  b2.y = B1[32];
        b3.x = B0[48];  b3.y = B1[48];

        acc0 = __builtin_amdgcn_wmma_f32_16x16x4_f32(false, a, false, b0,
                                                     (short)0, acc0, false, false);
        acc1 = __builtin_amdgcn_wmma_f32_16x16x4_f32(false, a, false, b1,
                                                     (short)0, acc1, false, false);
        acc2 = __builtin_amdgcn_wmma_f32_16x16x4_f32(false, a, false, b2,
                                                     (short)0, acc2, false, false);
        acc3 = __builtin_amdgcn_wmma_f32_16x16x4_f32(false, a, false, b3,
                                                     (short)0, acc3, false, false);
    }

    // C/D layout: lane L, vgpr v -> D[v + 8*(L>>4)][L&15]
    const int hi    = lane >> 4;
    const int cbase = col0 + m;
    const float bia0 = bias[cbase +  0];
    const float bia1 = bias[cbase + 16];
    const float bia2 = bias[cbase + 32];
    const float bia3 = bias[cbase + 48];

    #pragma unroll
    for (int v = 0; v < 8; ++v) {
        int r = row0 + v + 8 * hi;
        if (permute_rows) {
            // streaming output: non-temporal stores
            float* Crow = C + (long)(((r & 255) << 3) | (r >> 8)) * N + cbase;
            __builtin_nontemporal_store(acc0[v] + bia0, Crow +  0);
            __builtin_nontemporal_store(acc1[v] + bia1, Crow + 16);
            __builtin_nontemporal_store(acc2[v] + bia2, Crow + 32);
            __builtin_nontemporal_store(acc3[v] + bia3, Crow + 48);
        } else {
            float* Crow = C + (long)r * N + cbase;
            Crow[ 0] = acc0[v] + bia0;
            Crow[16] = acc1[v] + bia1;
            Crow[32] = acc2[v] + bia2;
            Crow[48] = acc3[v] + bia3;
        }
    }
}

// ---------------------------------------------------------------------------
// Sequential LSTM scan: 2048 steps, units = 512, gate width 4*512 = 2048.
// One persistent 1024-thread workgroup; h, c, z in LDS; per-step barrier.
// Each thread owns gate columns tid and tid+1024 of the h @ U GEMV
// (each wave reads 128B contiguous per U row; U stays L2-resident).
// x_seq row t maps to xz row (t & 255) because the sequence is merged
// tiled 8x.
// ---------------------------------------------------------------------------
__global__ __launch_bounds__(1024)
void lstm_scan_kernel(const float* __restrict__ xz,   // [256, 2048]
                      const float* __restrict__ U,    // [512, 2048]
                      float* __restrict__ hs) {       // [2048, 512]
    __shared__ float h[512];
    __shared__ float c[512];
    __shared__ float z[2048];
    const int tid = threadIdx.x;

    if (tid < 512) { h[tid] = 0.0f; c[tid] = 0.0f; }
    __syncthreads();

    const int j1 = tid;
    const int j2 = tid + 1024;

    for (int t = 0; t < 2048; ++t) {
        const float* xr = xz + (long)(t & 255) * 2048;
        float a1 = xr[j1];
        float a2 = xr[j2];
        #pragma unroll 8
        for (int k = 0; k < 512; ++k) {
            const float hk = h[k];
            a1 = fmaf(hk, U[(long)k * 2048 + j1], a1);
            a2 = fmaf(hk, U[(long)k * 2048 + j2], a2);
        }
        z[j1] = a1;
        z[j2] = a2;
        __syncthreads();   // all GEMV reads of h done; z visible

        if (tid < 512) {
            const float zi = z[tid];
            const float zf = z[tid + 512];
            const float zg = z[tid + 1024];
            const float zo = z[tid + 1536];
            const float ig = 1.0f / (1.0f + __expf(-zi));
            const float fg = 1.0f / (1.0f + __expf(-zf));
            const float gg = tanhf(zg);
            const float og = 1.0f / (1.0f + __expf(-zo));
            const float cn = fg * c[tid] + ig * gg;
            const float hn = og * tanhf(cn);
            c[tid] = cn;
            h[tid] = hn;
            hs[(long)t * 512 + tid] = hn;
        }
        __syncthreads();   // h/c updated before next step's GEMV
    }
}

// ---------------------------------------------------------------------------
// Inputs (setup_inputs order):
//   0 context [256,1024]  1 Wm [1024,512]  2 bm [512]
//   3 W [512,2048]        4 U [512,2048]   5 b [2048]
//   6 Wl [512,32000]      7 bl [32000]     8 max_subtoken_len (=8)
// Output: [256, 8, 32000] f32
// Workspace: merged 256x512 | xz 256x2048 | hs 2048x512   (~6.5 MB)
// ---------------------------------------------------------------------------
extern "C" void kernel_launch(void* const* d_in, const int* in_sizes, int n_in,
                              void* d_out, int out_size, void* d_ws, size_t ws_size,
                              hipStream_t stream) {
    const float* context = (const float*)d_in[0];
    const float* Wm      = (const float*)d_in[1];
    const float* bm      = (const float*)d_in[2];
    const float* W       = (const float*)d_in[3];
    const float* U       = (const float*)d_in[4];
    const float* b       = (const float*)d_in[5];
    const float* Wl      = (const float*)d_in[6];
    const float* bl      = (const float*)d_in[7];
    float* out = (float*)d_out;

    float* merged = (float*)d_ws;                 // 256*512
    float* xz     = merged + 256 * 512;           // 256*2048
    float* hs     = xz + 256 * 2048;              // 2048*512

    dim3 blk(256);

    // merged = context @ Wm + bm : M=256 N=512 K=1024 ; LDS = 16*1024*4 = 64 KB
    wmma_gemm_bias_f32<<<dim3(1, 16), blk, 16 * 1024 * 4, stream>>>(
        context, Wm, bm, merged, 256, 512, 1024, 0);

    // xz = merged @ W + b : M=256 N=2048 K=512 ; LDS = 32 KB
    wmma_gemm_bias_f32<<<dim3(4, 16), blk, 16 * 512 * 4, stream>>>(
        merged, W, b, xz, 256, 2048, 512, 0);

    // sequential recurrence over 2048 steps
    lstm_scan_kernel<<<1, 1024, 0, stream>>>(xz, U, hs);

    // logits = hs @ Wl + bl with (s,v)->(v,s) row permutation folded in:
    // M=2048 N=32000 K=512 ; grid.x = ceil(32000/512) = 63 ; LDS = 32 KB
    wmma_gemm_bias_f32<<<dim3(63, 128), blk, 16 * 512 * 4, stream>>>(
        hs, Wl, bl, out, 2048, 32000, 512, 1);
}